// Neural3DHMM_64115271795366
// MI455X (gfx1250) — compile-verified
//
#include <hip/hip_runtime.h>
#include <hip/hip_bf16.h>

// ---------------- problem constants (match reference) ----------------
#define BB 16       // batch
#define TT 128      // seq len
#define NS 1024     // states
#define VV 32000    // vocab

typedef float v2f __attribute__((ext_vector_type(2)));
typedef float v8f __attribute__((ext_vector_type(8)));

#define NEG_INF (-__builtin_inff())

// ============================================================================
// K1: per-row logsumexp of a (rows x V) matrix (emission rows; rows==1 reuse
// for the state-prior vector). One block per row; online LSE per thread,
// log-space tree combine in LDS.  Bandwidth-bound: 131 MB read @ 23.3 TB/s.
// ============================================================================
__global__ void k_row_lse(const float* __restrict__ E,
                          float* __restrict__ lse_out, int V) {
  __shared__ float sm[256];
  __shared__ float ss[256];
  const int row = blockIdx.x;
  const float* p = E + (size_t)row * (size_t)V;
  float m = NEG_INF, s = 0.0f;
  for (int v = threadIdx.x; v < V; v += 256) {
    float x = p[v];
    if (x > m) { s = s * __expf(m - x) + 1.0f; m = x; }
    else        s += __expf(x - m);
  }
  sm[threadIdx.x] = m; ss[threadIdx.x] = s;
  __syncthreads();
  for (int off = 128; off > 0; off >>= 1) {
    if (threadIdx.x < off) {
      float m1 = sm[threadIdx.x], s1 = ss[threadIdx.x];
      float m2 = sm[threadIdx.x + off], s2 = ss[threadIdx.x + off];
      float M = fmaxf(m1, m2);
      float S = (M == NEG_INF) ? 0.0f
                               : s1 * __expf(m1 - M) + s2 * __expf(m2 - M);
      sm[threadIdx.x] = M; ss[threadIdx.x] = S;
    }
    __syncthreads();
  }
  if (threadIdx.x == 0) lse_out[row] = sm[0] + __logf(ss[0]);
}

// ============================================================================
// K2: column logsumexp of transition (dim=0 softmax in reference).
// One thread per column; reads coalesced across threads (consecutive n).
// ============================================================================
__global__ void k_col_lse(const float* __restrict__ Tr,
                          float* __restrict__ colLSE) {
  const int n = blockIdx.x * 256 + threadIdx.x;   // 0..1023
  float m = NEG_INF, s = 0.0f;
  for (int r = 0; r < NS; ++r) {
    float x = Tr[(size_t)r * NS + n];
    if (x > m) { s = s * __expf(m - x) + 1.0f; m = x; }
    else        s += __expf(x - m);
  }
  colLSE[n] = m + __logf(s);
}

// ============================================================================
// K3: per-row max M_T[m] of log_T, then P[m,n] = exp(log_T[m,n] - M_T[m])
// written directly in WMMA A-fragment layout (16x4 f32):
//   frag index = ((tile*256 + kChunk)*32 + lane)*2 + j
//   lane 0-15  : M = tile*16+lane,    K = 4c + j      (j = 0,1)
//   lane 16-31 : M = tile*16+lane-16, K = 4c + 2 + j
// so the persistent kernel's A loads are contiguous 256B per wave.
// ============================================================================
__global__ void k_build_P(const float* __restrict__ Tr,
                          const float* __restrict__ colLSE,
                          float* __restrict__ MT,
                          float* __restrict__ Pfrag) {
  __shared__ float sred[256];
  __shared__ float sMT;
  const int m = blockIdx.x;
  const float* row = Tr + (size_t)m * NS;
  float mx = NEG_INF;
  for (int n = threadIdx.x; n < NS; n += 256)
    mx = fmaxf(mx, row[n] - colLSE[n]);
  sred[threadIdx.x] = mx;
  __syncthreads();
  for (int off = 128; off > 0; off >>= 1) {
    if (threadIdx.x < off)
      sred[threadIdx.x] = fmaxf(sred[threadIdx.x], sred[threadIdx.x + off]);
    __syncthreads();
  }
  if (threadIdx.x == 0) { sMT = sred[0]; MT[m] = sred[0]; }
  __syncthreads();
  const float MTv = sMT;

  const int c = threadIdx.x;          // k-chunk 0..255 (K = 4c..4c+3)
  const int tile = m >> 4, mr = m & 15;
  const int k = 4 * c;
  v2f lo, hi;
  lo.x = __expf(row[k + 0] - colLSE[k + 0] - MTv);
  lo.y = __expf(row[k + 1] - colLSE[k + 1] - MTv);
  hi.x = __expf(row[k + 2] - colLSE[k + 2] - MTv);
  hi.y = __expf(row[k + 3] - colLSE[k + 3] - MTv);
  v2f* base = (v2f*)Pfrag + ((size_t)tile * 256 + c) * 32;
  base[mr]      = lo;
  base[mr + 16] = hi;
}

// ============================================================================
// K4: pregather emissions in [t][m][b] layout with M_T (or log_pi at t=0)
// folded in:  emit_ws[t][m][b] = E[m, x[b,t]] - rowLSE[m] + (t? M_T[m]
//                                                              : pi[m]-piLSE)
// ============================================================================
__global__ void k_emit(const float* __restrict__ E,
                       const float* __restrict__ lse_row,
                       const float* __restrict__ MT,
                       const float* __restrict__ priors,
                       const float* __restrict__ pi_lse,
                       const int* __restrict__ tokens,
                       float* __restrict__ emit_ws) {
  const int bx = blockIdx.x;          // 0..B*T-1
  const int t = bx / BB;
  const int b = bx % BB;
  const int x = tokens[b * TT + t];
  const float* col = E + (size_t)x;
  const float plse = pi_lse[0];
#pragma unroll
  for (int i = 0; i < 4; ++i) {
    int mm = threadIdx.x + 256 * i;
    float base = (t == 0) ? (priors[mm] - plse) : MT[mm];
    emit_ws[((size_t)t * NS + mm) * BB + b] =
        col[(size_t)mm * VV] - lse_row[mm] + base;
  }
}

// ============================================================================
// K5: persistent forward recurrence.  1 workgroup, 1024 threads = 32 wave32.
// Wave w owns output tiles 2w, 2w+1 (16 states x 16 batch each).
// Per step:  per-b max/LSE reduction -> a_exp in LDS -> f32 WMMA GEMM
//            (M=1024, N=16, K=1024 via 16x16x4 f32) -> next alpha in regs.
// Emission values for t+1 are prefetched into registers BEFORE the GEMM so
// their L2 latency hides behind the 256-deep WMMA chain (serial path).
// ============================================================================
__global__ __launch_bounds__(1024)
void k_forward(const float* __restrict__ Pfrag,
               const float* __restrict__ emit_ws,
               const int* __restrict__ length,
               float* __restrict__ sums_ws,
               float* __restrict__ out) {
  __shared__ float s_aexp[NS * BB];   // [k][b] : exp(alpha - M_a[b])   64 KB
  __shared__ float s_red[32 * 16];    // per-wave partials               2 KB
  __shared__ float s_maprev[16];
  __shared__ float s_mnew[16];

  const int tid  = threadIdx.x;
  const int lane = tid & 31;
  const int wave = tid >> 5;          // 0..31
  const int half = lane >> 4;         // 0 | 1
  const int lb   = lane & 15;         // batch index of this lane
  const int tile0 = wave * 2, tile1 = wave * 2 + 1;

  float al[16];                       // this lane's alpha values (b = lb)

  // alpha_0 = emit[0] (+ log_pi folded in by K4)
#pragma unroll
  for (int j = 0; j < 8; ++j) {
    int m0 = tile0 * 16 + 8 * half + j;
    int m1 = tile1 * 16 + 8 * half + j;
    al[j]     = emit_ws[(size_t)m0 * BB + lb];
    al[8 + j] = emit_ws[(size_t)m1 * BB + lb];
  }

  for (int t = 0; t < TT; ++t) {
    // ---- reduce alpha_t: M_a[b], sums_log[b][t], build a_exp ----
    float lm = al[0];
#pragma unroll
    for (int j = 1; j < 16; ++j) lm = fmaxf(lm, al[j]);
    lm = fmaxf(lm, __shfl_xor(lm, 16));       // lanes l & l+16 share b
    if (half == 0) s_red[wave * 16 + lb] = lm;
    __syncthreads();                          // [A] (all GEMM reads done)
    if (tid < 16) {
      float M = s_red[tid];
#pragma unroll
      for (int w = 1; w < 32; ++w) M = fmaxf(M, s_red[w * 16 + tid]);
      s_mnew[tid] = M;
    }
    __syncthreads();                          // [B]
    const float Mb = s_mnew[lb];
    float ssum = 0.0f;
#pragma unroll
    for (int j = 0; j < 8; ++j) {
      int m0 = tile0 * 16 + 8 * half + j;
      int m1 = tile1 * 16 + 8 * half + j;
      float e0 = __expf(al[j] - Mb);
      float e1 = __expf(al[8 + j] - Mb);
      s_aexp[m0 * BB + lb] = e0;
      s_aexp[m1 * BB + lb] = e1;
      ssum += e0 + e1;
    }
    ssum += __shfl_xor(ssum, 16);
    if (half == 0) s_red[wave * 16 + lb] = ssum;
    __syncthreads();                          // [C]
    if (tid < 16) {
      float S = s_red[tid];
#pragma unroll
      for (int w = 1; w < 32; ++w) S += s_red[w * 16 + tid];
      sums_ws[t * BB + tid] = s_mnew[tid] + __logf(S);
      s_maprev[tid] = s_mnew[tid];
    }
    __syncthreads();                          // [D]

    if (t == TT - 1) break;

    // ---- prefetch emit[t+1] + M_a into regs (hides behind GEMM) ----
    const float* et = emit_ws + (size_t)(t + 1) * NS * BB;
    const float map = s_maprev[lb];
    float em[16];
#pragma unroll
    for (int j = 0; j < 8; ++j) {
      int m0 = tile0 * 16 + 8 * half + j;
      int m1 = tile1 * 16 + 8 * half + j;
      em[j]     = et[(size_t)m0 * BB + lb];
      em[8 + j] = et[(size_t)m1 * BB + lb];
    }

    // ---- GEMM: acc[m,b] = sum_k P[m,k] * a_exp[k,b]  (f32 WMMA) ----
    v8f acc0 = {}; v8f acc1 = {};
    const v2f* pf0 = (const v2f*)Pfrag + (size_t)tile0 * 256 * 32;
    const v2f* pf1 = (const v2f*)Pfrag + (size_t)tile1 * 256 * 32;
    for (int c = 0; c < 256; ++c) {
      const int k = 4 * c + 2 * half;         // B 4x16: K split across halves
      v2f bf;
      bf.x = s_aexp[(k + 0) * BB + lb];
      bf.y = s_aexp[(k + 1) * BB + lb];
      v2f a0 = pf0[c * 32 + lane];            // coalesced 256B/wave
      v2f a1 = pf1[c * 32 + lane];
      acc0 = __builtin_amdgcn_wmma_f32_16x16x4_f32(
          false, a0, false, bf, (short)0, acc0, false, false);
      acc1 = __builtin_amdgcn_wmma_f32_16x16x4_f32(
          false, a1, false, bf, (short)0, acc1, false, false);
    }

    // ---- alpha_{t+1} = emitMT[t+1][m][b] + M_a[b] + log(acc) ----
#pragma unroll
    for (int j = 0; j < 8; ++j) {
      al[j]     = em[j]     + map + __logf(acc0[j]);
      al[8 + j] = em[8 + j] + map + __logf(acc1[j]);
    }
  }

  __syncthreads();
  if (tid < 16) {                     // thread b wrote sums_ws[:, b] itself
    int idx = length[tid] - 1;
    idx = idx < 0 ? 0 : (idx > TT - 1 ? TT - 1 : idx);
    out[tid] = sums_ws[idx * BB + tid];
  }
}

// ============================================================================
// launch
// ============================================================================
extern "C" void kernel_launch(void* const* d_in, const int* in_sizes, int n_in,
                              void* d_out, int out_size, void* d_ws,
                              size_t ws_size, hipStream_t stream) {
  const float* E      = (const float*)d_in[0];   // (N, V)
  const float* Tr     = (const float*)d_in[1];   // (N, N)
  const float* priors = (const float*)d_in[2];   // (N,)
  const int*   tokens = (const int*)d_in[3];     // (B, T)
  const int*   length = (const int*)d_in[4];     // (B,)
  float* out = (float*)d_out;                    // (B, 1)

  char* w = (char*)d_ws;
  float* lse_row = (float*)(w + 0);                    //  4 KB
  float* colLSE  = (float*)(w + 4096);                 //  4 KB
  float* MT      = (float*)(w + 8192);                 //  4 KB
  float* pi_lse  = (float*)(w + 12288);                //  4 B
  float* sums_ws = (float*)(w + 16384);                //  8 KB (T x B)
  float* Pfrag   = (float*)(w + 32768);                //  4 MB (N x N)
  float* emit_ws = (float*)(w + 32768 + (size_t)NS * NS * 4);  // 8 MB (T,N,B)

  k_row_lse<<<NS, 256, 0, stream>>>(E, lse_row, VV);
  k_row_lse<<<1, 256, 0, stream>>>(priors, pi_lse, NS);
  k_col_lse<<<NS / 256, 256, 0, stream>>>(Tr, colLSE);
  k_build_P<<<NS, 256, 0, stream>>>(Tr, colLSE, MT, Pfrag);
  k_emit<<<BB * TT, 256, 0, stream>>>(E, lse_row, MT, priors, pi_lse, tokens,
                                      emit_ws);
  k_forward<<<1, 1024, 0, stream>>>(Pfrag, emit_ws, length, sums_ws, out);
}